// HAN_9689446220156
// MI455X (gfx1250) — compile-verified
//
#include <hip/hip_runtime.h>
#include <math.h>

#define NN 50000
#define HH 128
#define PP 3
#define EE 400000

typedef float v2f __attribute__((ext_vector_type(2)));
typedef float v8f __attribute__((ext_vector_type(8)));

__device__ __forceinline__ float wave_reduce_add(float p) {
#pragma unroll
  for (int o = 16; o >= 1; o >>= 1) p += __shfl_xor(p, o, 32);
  return p;
}

// ---------------------------------------------------------------------------
// GEMM: Y[R,128] = X[R,128] @ W[128,128] + bias (+ rowscale[row]*bias2) ; opt tanh
// One block = 8 waves; wave w computes the 16x16 tile (rows rt*16.., cols w*16..)
// via 32 chained v_wmma_f32_16x16x4_f32 ops (K = 128).
// ---------------------------------------------------------------------------
__global__ __launch_bounds__(256) void gemm128(
    const float* __restrict__ X, const float* __restrict__ W,
    const float* __restrict__ bias, float* __restrict__ Y, int rtiles,
    const float* __restrict__ bias2, const float* __restrict__ rowscale,
    int do_tanh) {
  const int wv = threadIdx.x >> 5, lane = threadIdx.x & 31;
  const int rt = blockIdx.x;
  if (rt >= rtiles) return;
  const int rowBase = rt << 4, colBase = wv << 4;
  const int m = lane & 15, hh = lane >> 4;
  v8f acc = {0.f, 0.f, 0.f, 0.f, 0.f, 0.f, 0.f, 0.f};
  const float* xr = X + (size_t)(rowBase + m) * HH;
#pragma unroll
  for (int kk = 0; kk < 32; ++kk) {
    const int k0 = (kk << 2) + (hh << 1);
    // A fragment (16x4 f32): lane holds A[m][k0], A[m][k0+1]
    v2f a;
    a.x = xr[k0];
    a.y = xr[k0 + 1];
    // B fragment (4x16 f32): lane holds B[k0][n], B[k0+1][n], n = lane&15
    v2f b;
    b.x = W[(size_t)k0 * HH + colBase + m];
    b.y = W[(size_t)(k0 + 1) * HH + colBase + m];
    acc = __builtin_amdgcn_wmma_f32_16x16x4_f32(false, a, false, b, (short)0,
                                                acc, false, false);
  }
#pragma unroll
  for (int r = 0; r < 8; ++r) {
    const int row = rowBase + r + (hh << 3);
    const int col = colBase + m;
    float y = acc[r] + bias[col];
    if (bias2) y += (rowscale ? rowscale[row] : 1.f) * bias2[col];
    if (do_tanh) y = tanhf(y);
    Y[(size_t)row * HH + col] = y;
  }
}

// ---------------------------------------------------------------------------
// outv[i] = dot(Mat[i,:], vec)  for i<128 ;  outv[128] = dot(gb, vec)
// one wave per output.
// ---------------------------------------------------------------------------
__global__ __launch_bounds__(256) void matvec129(
    const float* __restrict__ Mat, const float* __restrict__ gb,
    const float* __restrict__ vec, float* __restrict__ outv) {
  const int wv = (int)((blockIdx.x * (size_t)blockDim.x + threadIdx.x) >> 5);
  const int lane = threadIdx.x & 31;
  if (wv > 128) return;
  const float* row = (wv < 128) ? (Mat + (size_t)wv * HH) : gb;
  const float4 r = ((const float4*)row)[lane];
  const float4 v = ((const float4*)vec)[lane];
  float p = r.x * v.x + r.y * v.y + r.z * v.z + r.w * v.w;
  p = wave_reduce_add(p);
  if (lane == 0) outv[wv] = p;
}

// out[row] = dot(X[row,:], vc[0:128]) + vc[128]   (wave per row)
__global__ __launch_bounds__(256) void rowdot(const float* __restrict__ X,
                                              const float* __restrict__ vc,
                                              float* __restrict__ out, int R) {
  const int row = (int)((blockIdx.x * (size_t)blockDim.x + threadIdx.x) >> 5);
  const int lane = threadIdx.x & 31;
  if (row >= R) return;
  const float4 x = ((const float4*)(X + (size_t)row * HH))[lane];
  const float4 v = ((const float4*)vc)[lane];
  float p = x.x * v.x + x.y * v.y + x.z * v.z + x.w * v.w;
  p = wave_reduce_add(p);
  if (lane == 0) out[row] = p + vc[128];
}

// ew[e] = exp(tanh(a_s[si]+a_t[ti]+ab)); den[si] += ew[e]
__global__ void edge_pass1(const int* __restrict__ si, const int* __restrict__ ti,
                           const float* __restrict__ a_s,
                           const float* __restrict__ a_t,
                           const float* __restrict__ ab_p,
                           float* __restrict__ ew, float* __restrict__ den,
                           int E) {
  const int e = blockIdx.x * blockDim.x + threadIdx.x;
  if (e >= E) return;
  const int s = si[e], t = ti[e];
  const float w = __expf(tanhf(a_s[s] + a_t[t] + ab_p[0]));
  ew[e] = w;
  atomicAdd(&den[s], w);
}

__global__ void den_flag(const float* __restrict__ den, float* __restrict__ s,
                         int R) {
  const int i = blockIdx.x * blockDim.x + threadIdx.x;
  if (i < R) s[i] = den[i] > 0.f ? 1.f : 0.f;
}

// agg[si,:] += (ew[e]/den[si]) * tgt_h[ti,:]   (wave per edge, 4 f32/lane)
__global__ __launch_bounds__(256) void edge_pass2(
    const int* __restrict__ si, const int* __restrict__ ti,
    const float* __restrict__ ew, const float* __restrict__ den,
    const float* __restrict__ tgt, float* __restrict__ agg, int E) {
  const int e = (int)((blockIdx.x * (size_t)blockDim.x + threadIdx.x) >> 5);
  const int lane = threadIdx.x & 31;
  if (e >= E) return;
  const int s = si[e], t = ti[e];
  const float attn = ew[e] / den[s];
  const float4 th = ((const float4*)(tgt + (size_t)t * HH))[lane];
  float* dst = agg + (size_t)s * HH + lane * 4;
  atomicAdd(dst + 0, attn * th.x);
  atomicAdd(dst + 1, attn * th.y);
  atomicAdd(dst + 2, attn * th.z);
  atomicAdd(dst + 3, attn * th.w);
}

// semantic attention over 4 channels (src_h + 3 metapath embeddings)
__global__ __launch_bounds__(256) void semantic_attn(
    const float* __restrict__ srcH, const float* __restrict__ gat,
    const float* __restrict__ sem_w, const float* __restrict__ sem_b,
    float* __restrict__ out, int R) {
  const int row = (int)((blockIdx.x * (size_t)blockDim.x + threadIdx.x) >> 5);
  const int lane = threadIdx.x & 31;
  if (row >= R) return;
  const float4 wv = ((const float4*)sem_w)[lane];
  const float4 e0 = ((const float4*)(srcH + (size_t)row * HH))[lane];
  const float4 e1 =
      ((const float4*)(gat + 0 * (size_t)NN * HH + (size_t)row * HH))[lane];
  const float4 e2 =
      ((const float4*)(gat + 1 * (size_t)NN * HH + (size_t)row * HH))[lane];
  const float4 e3 =
      ((const float4*)(gat + 2 * (size_t)NN * HH + (size_t)row * HH))[lane];
  float p0 = e0.x * wv.x + e0.y * wv.y + e0.z * wv.z + e0.w * wv.w;
  float p1 = e1.x * wv.x + e1.y * wv.y + e1.z * wv.z + e1.w * wv.w;
  float p2 = e2.x * wv.x + e2.y * wv.y + e2.z * wv.z + e2.w * wv.w;
  float p3 = e3.x * wv.x + e3.y * wv.y + e3.z * wv.z + e3.w * wv.w;
  p0 = wave_reduce_add(p0);
  p1 = wave_reduce_add(p1);
  p2 = wave_reduce_add(p2);
  p3 = wave_reduce_add(p3);
  const float sb = sem_b[0];
  float s0 = p0 + sb, s1 = p1 + sb, s2 = p2 + sb, s3 = p3 + sb;
  s0 = s0 > 0.f ? s0 : 0.01f * s0;
  s1 = s1 > 0.f ? s1 : 0.01f * s1;
  s2 = s2 > 0.f ? s2 : 0.01f * s2;
  s3 = s3 > 0.f ? s3 : 0.01f * s3;
  const float mx = fmaxf(fmaxf(s0, s1), fmaxf(s2, s3));
  const float x0 = __expf(s0 - mx), x1 = __expf(s1 - mx), x2 = __expf(s2 - mx),
              x3 = __expf(s3 - mx);
  const float inv = 1.f / (x0 + x1 + x2 + x3);
  const float w0 = x0 * inv, w1 = x1 * inv, w2 = x2 * inv, w3 = x3 * inv;
  float4 o;
  o.x = w0 * e0.x + w1 * e1.x + w2 * e2.x + w3 * e3.x;
  o.y = w0 * e0.y + w1 * e1.y + w2 * e2.y + w3 * e3.y;
  o.z = w0 * e0.z + w1 * e1.z + w2 * e2.z + w3 * e3.z;
  o.w = w0 * e0.w + w1 * e1.w + w2 * e2.w + w3 * e3.w;
  ((float4*)(out + (size_t)row * HH))[lane] = o;
}

// LayerNorm of (in1 [+ in2]) with optional tanh first (wave per row)
__global__ __launch_bounds__(256) void ln128(
    const float* __restrict__ in1, const float* __restrict__ in2,
    const float* __restrict__ g, const float* __restrict__ beta,
    float* __restrict__ out, int R, int do_tanh) {
  const int row = (int)((blockIdx.x * (size_t)blockDim.x + threadIdx.x) >> 5);
  const int lane = threadIdx.x & 31;
  if (row >= R) return;
  float4 x = ((const float4*)(in1 + (size_t)row * HH))[lane];
  if (in2) {
    const float4 y = ((const float4*)(in2 + (size_t)row * HH))[lane];
    x.x += y.x; x.y += y.y; x.z += y.z; x.w += y.w;
  }
  if (do_tanh) {
    x.x = tanhf(x.x); x.y = tanhf(x.y); x.z = tanhf(x.z); x.w = tanhf(x.w);
  }
  float s = x.x + x.y + x.z + x.w;
  float q = x.x * x.x + x.y * x.y + x.z * x.z + x.w * x.w;
  s = wave_reduce_add(s);
  q = wave_reduce_add(q);
  const float mu = s * (1.f / HH);
  const float var = q * (1.f / HH) - mu * mu;
  const float rs = rsqrtf(var + 1e-5f);
  const float4 gv = ((const float4*)g)[lane];
  const float4 bv = ((const float4*)beta)[lane];
  float4 o;
  o.x = (x.x - mu) * rs * gv.x + bv.x;
  o.y = (x.y - mu) * rs * gv.y + bv.y;
  o.z = (x.z - mu) * rs * gv.z + bv.z;
  o.w = (x.w - mu) * rs * gv.w + bv.w;
  ((float4*)(out + (size_t)row * HH))[lane] = o;
}

// out[row] = sigmoid(dot(X[row,:], Wc) + bc)
__global__ __launch_bounds__(256) void cls128(const float* __restrict__ X,
                                              const float* __restrict__ Wc,
                                              const float* __restrict__ bc,
                                              float* __restrict__ out, int R) {
  const int row = (int)((blockIdx.x * (size_t)blockDim.x + threadIdx.x) >> 5);
  const int lane = threadIdx.x & 31;
  if (row >= R) return;
  const float4 x = ((const float4*)(X + (size_t)row * HH))[lane];
  const float4 w = ((const float4*)Wc)[lane];
  float p = x.x * w.x + x.y * w.y + x.z * w.z + x.w * w.w;
  p = wave_reduce_add(p);
  if (lane == 0) out[row] = 1.f / (1.f + __expf(-(p + bc[0])));
}

// ---------------------------------------------------------------------------
extern "C" void kernel_launch(void* const* d_in, const int* in_sizes, int n_in,
                              void* d_out, int out_size, void* d_ws,
                              size_t ws_size, hipStream_t stream) {
  (void)in_sizes; (void)n_in; (void)out_size; (void)ws_size;
  const float* src_x   = (const float*)d_in[0];
  const float* mp_x    = (const float*)d_in[1];
  const int*   edges   = (const int*)d_in[2];
  const float* W_src   = (const float*)d_in[3];
  const float* b_src   = (const float*)d_in[4];
  const float* W_tgt   = (const float*)d_in[5];
  const float* b_tgt   = (const float*)d_in[6];
  const float* gat_W   = (const float*)d_in[7];
  const float* gat_b   = (const float*)d_in[8];
  const float* att_w   = (const float*)d_in[9];
  const float* att_b   = (const float*)d_in[10];
  const float* gat_bias= (const float*)d_in[11];
  const float* sem_w   = (const float*)d_in[12];
  const float* sem_b   = (const float*)d_in[13];
  const float* prep_W  = (const float*)d_in[14];
  const float* prep_b  = (const float*)d_in[15];
  const float* dnn_W   = (const float*)d_in[16];
  const float* dnn_b   = (const float*)d_in[17];
  const float* dnn_g   = (const float*)d_in[18];
  const float* dnn_beta= (const float*)d_in[19];
  const float* res_g   = (const float*)d_in[20];
  const float* res_beta= (const float*)d_in[21];
  const float* cls_W   = (const float*)d_in[22];
  const float* cls_b   = (const float*)d_in[23];

  float* ws = (float*)d_ws;
  const size_t NHf = (size_t)NN * HH;
  float* bufA   = ws;                 // src_h           [N,H]
  float* bufT   = ws + NHf;           // tgt_h per path  [N,H]  (later: x3)
  float* bufAgg = ws + 2 * NHf;       // agg per path    [N,H]
  float* bufGat = ws + 3 * NHf;       // gat_out         [3,N,H] (later: hs,x1,x2)
  float* bufH   = ws + 6 * NHf;       // semantic h      [N,H]
  float* sm     = ws + 7 * NHf;
  float* a_s    = sm;                 // [N]
  float* a_t    = sm + NN;            // [N]
  float* den    = sm + 2 * NN;        // [N]
  float* sflag  = sm + 3 * NN;        // [N]
  float* ew     = sm + 4 * NN;        // [E]
  float* vs     = ew + EE;            // [132] (padded for float4 align)
  float* us     = vs + 132;           // [132]

  const int RT = NN / 16;             // 3125 row tiles
  const dim3 blk(256);
  const dim3 gGemm(RT);
  const dim3 gRow((NN + 7) / 8);      // wave-per-row kernels
  const dim3 gE1((EE + 255) / 256);
  const dim3 gE2((EE + 7) / 8);
  const dim3 gFlag((NN + 255) / 256);

  // 1. src_h = src_x @ W_src + b_src
  gemm128<<<gGemm, blk, 0, stream>>>(src_x, W_src, b_src, bufA, RT, nullptr,
                                     nullptr, 0);

  // 2. per-metapath GAT
  for (int p = 0; p < PP; ++p) {
    const float* gW = gat_W + (size_t)p * HH * HH;
    const float* gb = gat_b + (size_t)p * HH;
    const float* awp = att_w + (size_t)p * 2 * HH;
    gemm128<<<gGemm, blk, 0, stream>>>(mp_x + (size_t)p * NHf, W_tgt, b_tgt,
                                       bufT, RT, nullptr, nullptr, 0);
    matvec129<<<17, blk, 0, stream>>>(gW, gb, awp, vs);        // gW@aw_s , gb.aw_s
    matvec129<<<17, blk, 0, stream>>>(gW, gb, awp + HH, us);   // gW@aw_t , gb.aw_t
    rowdot<<<gRow, blk, 0, stream>>>(bufA, vs, a_s, NN);
    rowdot<<<gRow, blk, 0, stream>>>(bufT, us, a_t, NN);
    (void)hipMemsetAsync(den, 0, (size_t)NN * sizeof(float), stream);
    (void)hipMemsetAsync(bufAgg, 0, NHf * sizeof(float), stream);
    const int* si = edges + (size_t)p * 2 * EE;
    const int* ti = si + EE;
    edge_pass1<<<gE1, blk, 0, stream>>>(si, ti, a_s, a_t, att_b + p, ew, den, EE);
    den_flag<<<gFlag, blk, 0, stream>>>(den, sflag, NN);
    edge_pass2<<<gE2, blk, 0, stream>>>(si, ti, ew, den, bufT, bufAgg, EE);
    // gat_out = agg @ gW + gat_bias + sflag[row]*gb
    gemm128<<<gGemm, blk, 0, stream>>>(bufAgg, gW, gat_bias + (size_t)p * HH,
                                       bufGat + (size_t)p * NHf, RT, gb, sflag, 0);
  }

  // 3. semantic attention -> h
  semantic_attn<<<gRow, blk, 0, stream>>>(bufA, bufGat, sem_w, sem_b, bufH, NN);

  // 4. MLP head (reuse bufGat region: hs, x1, x2 ; x3 -> bufT)
  float* bufHS = bufGat;
  float* bufX1 = bufGat + NHf;
  float* bufX2 = bufGat + 2 * NHf;
  float* bufX3 = bufT;
  gemm128<<<gGemm, blk, 0, stream>>>(bufH, prep_W, prep_b, bufHS, RT, nullptr,
                                     nullptr, 0);
  gemm128<<<gGemm, blk, 0, stream>>>(bufHS, dnn_W, dnn_b, bufX1, RT, nullptr,
                                     nullptr, 1);
  ln128<<<gRow, blk, 0, stream>>>(bufX1, nullptr, dnn_g, dnn_beta, bufX1, NN, 0);
  gemm128<<<gGemm, blk, 0, stream>>>(bufX1, dnn_W + HH * HH, dnn_b + HH, bufX2,
                                     RT, nullptr, nullptr, 1);
  ln128<<<gRow, blk, 0, stream>>>(bufX2, nullptr, dnn_g + HH, dnn_beta + HH,
                                  bufX2, NN, 0);
  ln128<<<gRow, blk, 0, stream>>>(bufHS, bufX2, res_g, res_beta, bufX3, NN, 1);

  // 5. classifier -> sigmoid -> d_out [N]
  cls128<<<gRow, blk, 0, stream>>>(bufX3, cls_W, cls_b, (float*)d_out, NN);
}